// EdgeMatcher_3118146257456
// MI455X (gfx1250) — compile-verified
//
#include <hip/hip_runtime.h>
#include <math.h>

#define S_SAMPLES 1000
#define BIG 1e30f

// ---------------------------------------------------------------------------
// Kernel 1: one wave32 per edge point.
//  - lane l of tile k samples s = 32k + l (coalesced along the line)
//  - tile-boundary neighbors (s-1 for lane 0, s+1 for lane 31) come from one
//    extra EXEC-masked 2-lane gather, so every tile is fully independent and
//    the loop unrolls into many concurrent global_load_b32 gathers (high MLP)
//  - strict-local-minimum test uses wave32 shuffles for intra-tile neighbors
//  - per-lane best (dist, s, val, cx, cy) -> shfl_xor wave argmin reduction
// ---------------------------------------------------------------------------
__global__ void edge_minima_kernel(const float* __restrict__ pts,    // [N,2] (x,y)
                                   const float* __restrict__ nrms,   // [N,2]
                                   const float* __restrict__ img,    // [H,W]
                                   float* __restrict__ dist_out,     // [N,2] (d_out+1)
                                   float* __restrict__ sq_ws,        // [N]
                                   float* __restrict__ valid_ws,     // [N]
                                   int N, int W, int H)
{
    const int lane = threadIdx.x & 31;
    const int wave = (blockIdx.x * blockDim.x + threadIdx.x) >> 5;
    if (wave >= N) return;

    const float px = pts[2 * wave + 0];
    const float py = pts[2 * wave + 1];
    float nx = nrms[2 * wave + 0];
    float ny = nrms[2 * wave + 1];
    // reference re-normalizes: n = en / (||en|| + 1e-8)
    const float inv = 1.0f / (sqrtf(nx * nx + ny * ny) + 1e-8f);
    nx *= inv; ny *= inv;

    const float Wm1 = (float)(W - 1);
    const float Hm1 = (float)(H - 1);
    const float diag = sqrtf((float)(H * H + W * W));
    const float t0 = -0.5f * diag;
    const float dt = diag / (float)(S_SAMPLES - 1);

    // sample helper: image value at sample index s; outputs clipped coords
    auto sample = [&](int s, float& cx, float& cy) -> float {
        float t = t0 + (float)s * dt;
        float x = px + t * nx;
        float y = py + t * ny;
        cx = fminf(fmaxf(x, 0.0f), Wm1);
        cy = fminf(fmaxf(y, 0.0f), Hm1);
        int xi = (int)rintf(cx);           // round-half-to-even == jnp.round
        int yi = (int)rintf(cy);
        xi = min(max(xi, 0), W - 1);
        yi = min(max(yi, 0), H - 1);
        return img[(size_t)yi * (size_t)W + (size_t)xi];
    };

    // per-lane best tuple (lexicographic min over (dist, s) -> order independent)
    float bd  = BIG;          // best masked distance
    int   bs  = 0x7fffffff;   // best sample index (tie-break: smallest s)
    float bv  = 0.0f;         // value at best minimum
    float bcx = 0.0f, bcy = 0.0f;

    // --- 32 fully independent tiles of 32 samples (s in [0,1023]; s>998 masked)
#pragma unroll 4
    for (int k = 0; k < 32; ++k) {
        const int s = k * 32 + lane;

        float cx, cy;
        const float v = sample(s, cx, cy);

        // boundary neighbor: lane 0 needs v(32k-1), lane 31 needs v(32k+32).
        // EXEC-masked load -> only 2 memory requests per tile.
        float vb = 0.0f;
        if (lane == 0 || lane == 31) {
            int sb = (lane == 0) ? (k * 32 - 1) : (k * 32 + 32);
            sb = max(sb, 0);                 // s=-1 only when k==0/lane 0 (masked)
            float tx, ty;
            vb = sample(sb, tx, ty);
        }

        const float vup = __shfl_up(v, 1, 32);
        const float vdn = __shfl_down(v, 1, 32);
        const float vprev = (lane == 0)  ? vb : vup;
        const float vnext = (lane == 31) ? vb : vdn;

        const bool is_min = (s >= 1) && (s <= S_SAMPLES - 2) &&
                            (v < vprev) && (v < vnext);
        if (is_min) {
            const float dx = cx - px, dy = cy - py;
            const float d = sqrtf(dx * dx + dy * dy);
            if (d < bd || (d == bd && s < bs)) {
                bd = d; bs = s; bv = v; bcx = cx; bcy = cy;
            }
        }
    }

    // --- wave32 reduction of best tuples (argmin by dist, then smallest s)
    for (int off = 16; off > 0; off >>= 1) {
        float od = __shfl_xor(bd,  off, 32);
        int   os = __shfl_xor(bs,  off, 32);
        float ov = __shfl_xor(bv,  off, 32);
        float ox = __shfl_xor(bcx, off, 32);
        float oy = __shfl_xor(bcy, off, 32);
        if (od < bd || (od == bd && os < bs)) {
            bd = od; bs = os; bv = ov; bcx = ox; bcy = oy;
        }
    }

    if (lane == 0) {
        const bool valid = (bd < 1e29f);

        // reference value at the edge point itself (round + clip)
        const int rxi = min(max((int)rintf(px), 0), W - 1);
        const int ryi = min(max((int)rintf(py), 0), H - 1);
        const float refv = img[(size_t)ryi * (size_t)W + (size_t)rxi];

        dist_out[2 * wave + 0] = valid ? (bcx - px) : 0.0f;
        dist_out[2 * wave + 1] = valid ? (bcy - py) : 0.0f;

        const float diff = refv - bv;
        sq_ws[wave]    = valid ? (diff * diff) : 0.0f;
        valid_ws[wave] = valid ? 1.0f : 0.0f;
    }
}

// ---------------------------------------------------------------------------
// Kernel 2: deterministic fixed-order reduction of sq/valid -> loss scalar.
// ---------------------------------------------------------------------------
__global__ void edge_loss_reduce_kernel(const float* __restrict__ sq_ws,
                                        const float* __restrict__ valid_ws,
                                        float* __restrict__ loss_out, int N)
{
    __shared__ float ssq[256];
    __shared__ float sva[256];
    float a = 0.0f, b = 0.0f;
    for (int i = threadIdx.x; i < N; i += 256) {
        a += sq_ws[i];
        b += valid_ws[i];
    }
    ssq[threadIdx.x] = a;
    sva[threadIdx.x] = b;
    __syncthreads();
    for (int off = 128; off > 0; off >>= 1) {
        if ((int)threadIdx.x < off) {
            ssq[threadIdx.x] += ssq[threadIdx.x + off];
            sva[threadIdx.x] += sva[threadIdx.x + off];
        }
        __syncthreads();
    }
    if (threadIdx.x == 0) {
        loss_out[0] = ssq[0] / fmaxf(sva[0], 1.0f);
    }
}

extern "C" void kernel_launch(void* const* d_in, const int* in_sizes, int n_in,
                              void* d_out, int out_size, void* d_ws, size_t ws_size,
                              hipStream_t stream) {
    const float* pts  = (const float*)d_in[0];   // edge_points  [N,2]
    const float* nrms = (const float*)d_in[1];   // edge_normals [N,2]
    const float* img  = (const float*)d_in[2];   // distance_image [1,1,H,W]

    const int N  = in_sizes[0] / 2;
    const int HW = in_sizes[2];
    int W = (int)lroundf(sqrtf((float)HW));      // square image per reference setup
    int H = (W > 0) ? (HW / W) : 0;

    float* out      = (float*)d_out;             // [0]=loss, [1..2N]=distances
    float* sq_ws    = (float*)d_ws;              // [N]
    float* valid_ws = sq_ws + N;                 // [N]

    const int block = 256;                       // 8 wave32 per block
    const int wpb   = block / 32;
    const int grid  = (N + wpb - 1) / wpb;

    edge_minima_kernel<<<grid, block, 0, stream>>>(pts, nrms, img,
                                                   out + 1, sq_ws, valid_ws,
                                                   N, W, H);
    edge_loss_reduce_kernel<<<1, 256, 0, stream>>>(sq_ws, valid_ws, out, N);
}